// PQCRegressor_49486613184840
// MI455X (gfx1250) — compile-verified
//
#include <hip/hip_runtime.h>
#include <math.h>

typedef __attribute__((ext_vector_type(2))) float v2f;
typedef __attribute__((ext_vector_type(8))) float v8f;

#define XS_STRIDE 65   // 16x64 x-tile, padded (stride%64==1 -> conflict-free)
#define H_STRIDE  257  // 16x256 hidden tiles, padded

// ---------------------------------------------------------------------------
// Kernel 1: fused 3-layer MLP with V_WMMA_F32_16X16X4_F32.
// grid = B/16 blocks, 128 threads (4 waves). Each block owns 16 rows.
// A-fragment (f32 16x4): lane l holds A[M=l&15][K = k0 + 2*(l>>4) + v]
// B-fragment (f32 4x16): lane l holds B[K = k0 + 2*(l>>4) + v][N = l&15]
// C/D (f32 16x16):       lane l, vgpr v = C[M = v + 8*(l>>4)][N = l&15]
// ---------------------------------------------------------------------------
__global__ __launch_bounds__(128) void mlp_wmma_kernel(
    const float* __restrict__ x,
    const float* __restrict__ W1, const float* __restrict__ b1,
    const float* __restrict__ W2, const float* __restrict__ b2,
    const float* __restrict__ W3, const float* __restrict__ b3,
    float* __restrict__ params)
{
  __shared__ float xs[16 * XS_STRIDE];
  __shared__ float h1[16 * H_STRIDE];
  __shared__ float h2[16 * H_STRIDE];

  const int r0   = blockIdx.x * 16;
  const int tid  = threadIdx.x;
  const int lane = tid & 31;
  const int wid  = tid >> 5;
  const int m16  = lane & 15;   // M (A-frag row / N column selector)
  const int hl   = lane >> 4;   // half-wave selector

  // cooperative load of x tile [16 x 64]
  for (int idx = tid; idx < 16 * 64; idx += 128) {
    int i = idx >> 6, j = idx & 63;
    xs[i * XS_STRIDE + j] = x[(r0 + i) * 64 + j];
  }
  __syncthreads();

  // ---- layer 1: h1 = relu(x @ W1 + b1)   K=64, N=256 ----
  for (int nt = wid; nt < 16; nt += 4) {
    const int n = nt * 16 + m16;
    v8f acc = {0.f, 0.f, 0.f, 0.f, 0.f, 0.f, 0.f, 0.f};
    for (int k0 = 0; k0 < 64; k0 += 4) {
      const int kk = k0 + 2 * hl;
      v2f a, bf;
      a.x  = xs[m16 * XS_STRIDE + kk];
      a.y  = xs[m16 * XS_STRIDE + kk + 1];
      bf.x = W1[kk * 256 + n];
      bf.y = W1[(kk + 1) * 256 + n];
      acc = __builtin_amdgcn_wmma_f32_16x16x4_f32(false, a, false, bf,
                                                  (short)0, acc, false, false);
    }
    const float bias = b1[n];
#pragma unroll
    for (int v = 0; v < 8; ++v) {
      const int m = v + 8 * hl;
      const float val = acc[v] + bias;
      h1[m * H_STRIDE + n] = val > 0.f ? val : 0.f;
    }
  }
  __syncthreads();

  // ---- layer 2: h2 = relu(h1 @ W2 + b2)   K=256, N=256 ----
  for (int nt = wid; nt < 16; nt += 4) {
    const int n = nt * 16 + m16;
    v8f acc = {0.f, 0.f, 0.f, 0.f, 0.f, 0.f, 0.f, 0.f};
    for (int k0 = 0; k0 < 256; k0 += 4) {
      const int kk = k0 + 2 * hl;
      v2f a, bf;
      a.x  = h1[m16 * H_STRIDE + kk];
      a.y  = h1[m16 * H_STRIDE + kk + 1];
      bf.x = W2[kk * 256 + n];
      bf.y = W2[(kk + 1) * 256 + n];
      acc = __builtin_amdgcn_wmma_f32_16x16x4_f32(false, a, false, bf,
                                                  (short)0, acc, false, false);
    }
    const float bias = b2[n];
#pragma unroll
    for (int v = 0; v < 8; ++v) {
      const int m = v + 8 * hl;
      const float val = acc[v] + bias;
      h2[m * H_STRIDE + n] = val > 0.f ? val : 0.f;
    }
  }
  __syncthreads();

  // ---- layer 3: params = h2 @ W3 + b3   K=256, N=32 (waves 0,1 only) ----
  for (int nt = wid; nt < 2; nt += 4) {
    const int n = nt * 16 + m16;
    v8f acc = {0.f, 0.f, 0.f, 0.f, 0.f, 0.f, 0.f, 0.f};
    for (int k0 = 0; k0 < 256; k0 += 4) {
      const int kk = k0 + 2 * hl;
      v2f a, bf;
      a.x  = h2[m16 * H_STRIDE + kk];
      a.y  = h2[m16 * H_STRIDE + kk + 1];
      bf.x = W3[kk * 32 + n];
      bf.y = W3[(kk + 1) * 32 + n];
      acc = __builtin_amdgcn_wmma_f32_16x16x4_f32(false, a, false, bf,
                                                  (short)0, acc, false, false);
    }
    const float bias = b3[n];
#pragma unroll
    for (int v = 0; v < 8; ++v) {
      const int m = v + 8 * hl;
      params[(r0 + m) * 32 + n] = acc[v] + bias;
    }
  }
}

// ---------------------------------------------------------------------------
// Kernel 2: product-state expansion. state[c] = T[c ^ (c>>1)] where
// T[a] = prod_w q[w][bit_{15-w}(a)] (wire 0 = MSB). Split a into high/low
// bytes: T[a] = Ah[a>>8] * Bl[a&255], both 256-entry prefix products in LDS.
// 1 block per batch element, 256 threads, each emits 256 float2 NT stores.
// ---------------------------------------------------------------------------
__device__ inline v2f cmul(v2f p, v2f q) {
  v2f r;
  r.x = p.x * q.x - p.y * q.y;
  r.y = p.x * q.y + p.y * q.x;
  return r;
}

__global__ __launch_bounds__(256) void circuit_kernel(
    const float* __restrict__ params, float* __restrict__ out)
{
  __shared__ v2f q[32];     // q[2*w + bit]
  __shared__ v2f Ah[256];   // wires 0..7  (a bits 15..8)
  __shared__ v2f Bl[256];   // wires 8..15 (a bits 7..0)

  const int b = blockIdx.x;
  const int t = threadIdx.x;

  if (t < 16) {
    const float theta = 0.5f * params[b * 32 + t];
    const float phi   = 0.5f * params[b * 32 + 16 + t];
    float st, ct, sp, cp;
    sincosf(theta, &st, &ct);
    sincosf(phi, &sp, &cp);
    v2f a0, a1;
    a0.x = ct * cp;  a0.y = -ct * sp;   // cos(th) * exp(-i phi)
    a1.x = st * cp;  a1.y =  st * sp;   // sin(th) * exp(+i phi)
    q[2 * t]     = a0;
    q[2 * t + 1] = a1;
  }
  __syncthreads();

  // prefix products over each half of the wires
  v2f pa = q[(t >> 7) & 1];                       // wire 0 -> bit 7 of t
#pragma unroll
  for (int w = 1; w < 8; ++w)
    pa = cmul(pa, q[2 * w + ((t >> (7 - w)) & 1)]);
  Ah[t] = pa;

  v2f pb = q[16 + ((t >> 7) & 1)];                // wire 8 -> bit 7 of t
#pragma unroll
  for (int w = 1; w < 8; ++w)
    pb = cmul(pb, q[16 + 2 * w + ((t >> (7 - w)) & 1)]);
  Bl[t] = pb;
  __syncthreads();

  v2f* __restrict__ outp = reinterpret_cast<v2f*>(out) + (size_t)b * 65536;
  for (int k = 0; k < 256; ++k) {
    const int c = (k << 8) | t;                    // lane-consecutive: coalesced
    const int a = c ^ (c >> 1);                    // Gray permutation (CNOT chain)
    const v2f hi = Ah[a >> 8];                     // wave-uniform broadcast
    const v2f lo = Bl[a & 255];
    v2f val;
    val.x = hi.x * lo.x - hi.y * lo.y;
    val.y = hi.x * lo.y + hi.y * lo.x;
    __builtin_nontemporal_store(val, outp + c);    // 512MB stream: bypass temporal caching
  }
}

// ---------------------------------------------------------------------------
extern "C" void kernel_launch(void* const* d_in, const int* in_sizes, int n_in,
                              void* d_out, int out_size, void* d_ws, size_t ws_size,
                              hipStream_t stream) {
  const float* x  = (const float*)d_in[0];
  const float* W1 = (const float*)d_in[1];
  const float* b1 = (const float*)d_in[2];
  const float* W2 = (const float*)d_in[3];
  const float* b2 = (const float*)d_in[4];
  const float* W3 = (const float*)d_in[5];
  const float* b3 = (const float*)d_in[6];

  const int B = in_sizes[0] / 64;       // x is [B, 64]
  float* params = (float*)d_ws;         // [B, 32] scratch

  mlp_wmma_kernel<<<B / 16, 128, 0, stream>>>(x, W1, b1, W2, b2, W3, b3, params);
  circuit_kernel<<<B, 256, 0, stream>>>(params, (float*)d_out);
}